// TSBarrierModel_40836549050528
// MI455X (gfx1250) — compile-verified
//
#include <hip/hip_runtime.h>
#include <math.h>

// ---------------------------------------------------------------------------
// Workspace (d_ws, floats): [0..26] = C1 (3x3x3), [27..151] = C2 (5x5x5)
// d_out layout: [0..G) graph sums, [G .. G+10*E) edge embedding
// ---------------------------------------------------------------------------

static __device__ __forceinline__ int imax2(int a, int b) { return a > b ? a : b; }
static __device__ __forceinline__ int imin2(int a, int b) { return a < b ? a : b; }

static __device__ __forceinline__ double dfact(int n) {
  double r = 1.0;
  for (int i = 2; i <= n; ++i) r *= (double)i;
  return r;
}

static __device__ double cg_coeff(int j1, int m1, int j2, int m2, int j3, int m3) {
  if (m1 + m2 != m3) return 0.0;
  double pre = sqrt((2.0 * j3 + 1.0) * dfact(j3 + j1 - j2) * dfact(j3 - j1 + j2) *
                    dfact(j1 + j2 - j3) / dfact(j1 + j2 + j3 + 1));
  pre *= sqrt(dfact(j3 + m3) * dfact(j3 - m3) * dfact(j1 - m1) * dfact(j1 + m1) *
              dfact(j2 - m2) * dfact(j2 + m2));
  int kmin = imax2(0, imax2(j2 - j3 - m1, j1 - j3 + m2));
  int kmax = imin2(j1 + j2 - j3, imin2(j1 - m1, j2 + m2));
  double s = 0.0;
  for (int k = kmin; k <= kmax; ++k) {
    double d = dfact(k) * dfact(j1 + j2 - j3 - k) * dfact(j1 - m1 - k) *
               dfact(j2 + m2 - k) * dfact(j3 - j2 + m1 + k) * dfact(j3 - j1 - m2 + k);
    s += ((k & 1) ? -1.0 : 1.0) / d;
  }
  return pre * s;
}

static __device__ __forceinline__ void U_real(int l, int row, int col, double& re, double& im) {
  re = 0.0; im = 0.0;
  const double is2 = 0.70710678118654752440;
  if (row == l) {
    if (col == l) re = 1.0;
  } else if (row > l) {
    int m = row - l;
    if (col == l + m)      re = ((m & 1) ? -1.0 : 1.0) * is2;
    else if (col == l - m) re = is2;
  } else {
    int m = l - row;
    if (col == l - m)      im = is2;
    else if (col == l + m) im = -((m & 1) ? -1.0 : 1.0) * is2;
  }
}

// Blocks 0,1: CG tensors (l=1 -> ws[0..26], l=2 -> ws[27..151]).
// Blocks >=2: zero the G graph accumulators.
__global__ void cg_setup_kernel(float* __restrict__ C, float* __restrict__ out, int G) {
  if (blockIdx.x >= 2) {
    int i = (blockIdx.x - 2) * 128 + threadIdx.x;
    if (i < G) out[i] = 0.0f;
    return;
  }

  const int l = (blockIdx.x == 0) ? 1 : 2;
  const int n = 2 * l + 1;
  const int nn = n * n;
  const int total = nn * n;
  const int off = (blockIdx.x == 0) ? 0 : 27;
  const int tid = threadIdx.x;

  __shared__ double cgt[125];
  if (tid < total) {
    int i = tid / nn, j = (tid / n) % n, k = tid % n;
    cgt[tid] = cg_coeff(l, i - l, l, j - l, l, k - l);
  }
  __syncthreads();

  double re = 0.0, im = 0.0;
  if (tid < total) {
    int a = tid / nn, b = (tid / n) % n, c = tid % n;
    for (int i = 0; i < n; ++i) {
      double ar, ai;
      U_real(l, a, i, ar, ai);
      if (ar == 0.0 && ai == 0.0) continue;
      for (int j = 0; j < n; ++j) {
        double br, bi;
        U_real(l, b, j, br, bi);
        if (br == 0.0 && bi == 0.0) continue;
        double t1r = ar * br - ai * bi;
        double t1i = ar * bi + ai * br;
        for (int k = 0; k < n; ++k) {
          double w = cgt[(i * n + j) * n + k];
          if (w == 0.0) continue;
          double cr, ci;
          U_real(l, c, k, cr, ci);
          re += (t1r * cr + t1i * ci) * w;   // * conj(U[c,k])
          im += (t1i * cr - t1r * ci) * w;
        }
      }
    }
  }

  __shared__ double sd[128];
  sd[tid] = (tid < total) ? fabs(re) : 0.0; __syncthreads();
  for (int s = 64; s > 0; s >>= 1) { if (tid < s) sd[tid] = fmax(sd[tid], sd[tid + s]); __syncthreads(); }
  double maxRe = sd[0]; __syncthreads();
  sd[tid] = (tid < total) ? fabs(im) : 0.0; __syncthreads();
  for (int s = 64; s > 0; s >>= 1) { if (tid < s) sd[tid] = fmax(sd[tid], sd[tid + s]); __syncthreads(); }
  double maxIm = sd[0]; __syncthreads();

  double sel = (maxIm > maxRe) ? im : re;   // mirrors: if |imag| dominates, Cr/=1j; Cr=Cr.real
  sd[tid] = (tid < total) ? sel * sel : 0.0; __syncthreads();
  for (int s = 64; s > 0; s >>= 1) { if (tid < s) sd[tid] += sd[tid + s]; __syncthreads(); }
  double nrm = sqrt(sd[0]);

  if (tid < total) C[off + tid] = (float)(sel / nrm);
}

// ---------------------------------------------------------------------------
// Fused compute kernel:
//   blocks [0, atomBlocks)            -> per-atom features + atomic segment sum
//   blocks [atomBlocks, atomBlocks+eb) -> edge embedding (overlaps with atoms)
//
// Atom path exploits CG (anti)symmetry: l=1 coupling is antisymmetric in
// (i,j), l=2 symmetric (both preserved by the real-basis rotation), and
// sum over the full Gram == ||column sums of t||^2. Shell 0 (identity t)
// constant-folds under full unroll. C reads are wave-uniform -> scalar loads.
// H streams read-once -> non-temporal (keeps `pos` hot in L2 for edge blocks).
//
// Edge path: at most 2 of the 10 bump windows are active for any r, so only
// 2 rcp+exp (TRANS) pairs are evaluated; results are placed with unrolled
// compare-selects. Output staged in LDS (stride-10 writes: conflict-free for
// 32 lanes), streamed out as contiguous b128 non-temporal stores.
// ---------------------------------------------------------------------------
typedef float v2f __attribute__((ext_vector_type(2)));
typedef float v4f __attribute__((ext_vector_type(4)));

__global__ __launch_bounds__(256) void fused_kernel(
    const float* __restrict__ Hs, const float* __restrict__ Hp,
    const float* __restrict__ Hd, const float* __restrict__ pws,
    const float* __restrict__ pwp, const float* __restrict__ pwd,
    const int* __restrict__ gids, const float* __restrict__ C,
    float* __restrict__ out,
    const int* __restrict__ esrc, const int* __restrict__ edst,
    const float* __restrict__ pos, float* __restrict__ emb,
    int N, int E, int atomBlocks) {
  __shared__ float tile[2560];

  if ((int)blockIdx.x < atomBlocks) {
    // ===================== atom path =====================
    int n = blockIdx.x * 256 + threadIdx.x;
    if (n >= N) return;

    const float w_s = pws[0], w_p = pwp[0], w_d = pwd[0];

    // s channel: ts = prod(w_s * H_s); feat = ts^2 / 16
    v4f hsv = __builtin_nontemporal_load((const v4f*)(Hs + (size_t)4 * n));  // 16B aligned
    float ts = (w_s * hsv[0]) * (w_s * hsv[1]) * (w_s * hsv[2]) * (w_s * hsv[3]);
    float acc = ts * ts * (1.0f / 16.0f);

    // p channel: dim=3, 3 shells, antisymmetric C1, /9
    {
      const float scale = w_p * 1.7320508075688772f;  // sqrt(3)
      const float* H = Hp + (size_t)27 * n;
      float hp[27];
#pragma unroll
      for (int q = 0; q < 27; ++q) hp[q] = __builtin_nontemporal_load(H + q);

      float t[3][3];
#pragma unroll
      for (int b = 0; b < 3; ++b)
#pragma unroll
        for (int i = 0; i < 3; ++i) t[b][i] = (i == b) ? 1.0f : 0.0f;

#pragma unroll
      for (int sh = 0; sh < 3; ++sh) {
#pragma unroll
        for (int b = 0; b < 3; ++b) {
          float h0 = hp[sh * 9 + b * 3 + 0];
          float h1 = hp[sh * 9 + b * 3 + 1];
          float h2 = hp[sh * 9 + b * 3 + 2];
          float hb[3] = {h0, h1, h2};
          float nt[3] = {0.0f, 0.0f, 0.0f};
#pragma unroll
          for (int i = 0; i < 3; ++i)
#pragma unroll
            for (int j = i + 1; j < 3; ++j) {
              float pv = t[b][i] * hb[j] - t[b][j] * hb[i];
#pragma unroll
              for (int k = 0; k < 3; ++k) nt[k] = fmaf(C[(i * 3 + j) * 3 + k], pv, nt[k]);
            }
#pragma unroll
          for (int k = 0; k < 3; ++k) t[b][k] = scale * nt[k];
        }
      }
      float u0 = t[0][0] + t[1][0] + t[2][0];
      float u1 = t[0][1] + t[1][1] + t[2][1];
      float u2 = t[0][2] + t[1][2] + t[2][2];
      acc += (u0 * u0 + u1 * u1 + u2 * u2) * (1.0f / 9.0f);
    }

    // d channel: dim=5, 2 shells, symmetric C2, /4
    {
      const float scale = w_d * 2.2360679774997896f;  // sqrt(5)
      const float* H = Hd + (size_t)50 * n;           // 8B aligned (200n bytes)
      const float* C2 = C + 27;
      float hd[50];
#pragma unroll
      for (int q = 0; q < 25; ++q) {                  // 25 x b64 NT loads
        v2f v = __builtin_nontemporal_load((const v2f*)(H + 2 * q));
        hd[2 * q] = v[0];
        hd[2 * q + 1] = v[1];
      }

      float t[5][5];
#pragma unroll
      for (int b = 0; b < 5; ++b)
#pragma unroll
        for (int i = 0; i < 5; ++i) t[b][i] = (i == b) ? 1.0f : 0.0f;

#pragma unroll
      for (int sh = 0; sh < 2; ++sh) {
#pragma unroll
        for (int b = 0; b < 5; ++b) {
          float hb[5];
#pragma unroll
          for (int j = 0; j < 5; ++j) hb[j] = hd[sh * 25 + b * 5 + j];
          float nt[5] = {0.0f, 0.0f, 0.0f, 0.0f, 0.0f};
#pragma unroll
          for (int i = 0; i < 5; ++i) {                 // diagonal
            float pv = t[b][i] * hb[i];
#pragma unroll
            for (int k = 0; k < 5; ++k) nt[k] = fmaf(C2[(i * 5 + i) * 5 + k], pv, nt[k]);
          }
#pragma unroll
          for (int i = 0; i < 5; ++i)                   // upper triangle
#pragma unroll
            for (int j = i + 1; j < 5; ++j) {
              float pv = fmaf(t[b][i], hb[j], t[b][j] * hb[i]);
#pragma unroll
              for (int k = 0; k < 5; ++k) nt[k] = fmaf(C2[(i * 5 + j) * 5 + k], pv, nt[k]);
            }
#pragma unroll
          for (int k = 0; k < 5; ++k) t[b][k] = scale * nt[k];
        }
      }
      float u[5];
#pragma unroll
      for (int k = 0; k < 5; ++k)
        u[k] = t[0][k] + t[1][k] + t[2][k] + t[3][k] + t[4][k];
      acc += (u[0] * u[0] + u[1] * u[1] + u[2] * u[2] + u[3] * u[3] + u[4] * u[4]) * 0.25f;
    }

    int g = __builtin_nontemporal_load(gids + n);
    unsafeAtomicAdd(&out[g], acc);   // hardware global_atomic_add_f32
    return;
  }

  // ===================== edge path =====================
  const int base = ((int)blockIdx.x - atomBlocks) * 256;
  const int e = base + threadIdx.x;

  float o[10];
#pragma unroll
  for (int q = 0; q < 10; ++q) o[q] = 0.0f;

  if (e < E) {
    int s = __builtin_nontemporal_load(esrc + e);
    int d = __builtin_nontemporal_load(edst + e);

    float dx = pos[3 * d + 0] - pos[3 * s + 0];
    float dy = pos[3 * d + 1] - pos[3 * s + 1];
    float dz = pos[3 * d + 2] - pos[3 * s + 2];
    float r = sqrtf(fmaf(dx, dx, fmaf(dy, dy, fmaf(dz, dz, 1e-12f))));

    // d_q = a - (q+1), a = r*(11/5); active windows: q+1 in {floor(a), floor(a)+1}
    const float a = r * (11.0f / 5.0f);
    const float K = 1.14136f * 3.16227766016838f;  // bump coeff * sqrt(NUM_BASIS)

    float f = floorf(a);
    float v1 = a - f;           // in [0,1): inside window m1 = f
    float v2 = v1 - 1.0f;       // in [-1,0): inside window m2 = f+1 (den2=0 -> exp(-inf)=0)
    float den1 = 1.0f - v1 * v1;
    float den2 = 1.0f - v2 * v2;
    float val1 = K * __expf(2.0f - 2.0f * __builtin_amdgcn_rcpf(den1));
    float val2 = K * __expf(2.0f - 2.0f * __builtin_amdgcn_rcpf(den2));
    int m1 = (int)f;            // a >= 0 always
    int m2 = m1 + 1;

#pragma unroll
    for (int q = 0; q < 10; ++q) {
      float res = (q + 1 == m1) ? val1 : 0.0f;
      o[q] = (q + 1 == m2) ? val2 : res;
    }
  }

#pragma unroll
  for (int q = 0; q < 10; ++q) tile[threadIdx.x * 10 + q] = o[q];
  __syncthreads();

  float* gbase = emb + (size_t)base * 10;          // 16B aligned (base*40 bytes)
  const int limit = imin2(2560, 10 * (E - base));  // floats belonging to valid edges
  for (int i = threadIdx.x * 4; i < limit; i += 1024) {
    v4f v = *(const v4f*)&tile[i];
    __builtin_nontemporal_store(v, (v4f*)(gbase + i));
  }
}

// ---------------------------------------------------------------------------
extern "C" void kernel_launch(void* const* d_in, const int* in_sizes, int n_in,
                              void* d_out, int out_size, void* d_ws, size_t ws_size,
                              hipStream_t stream) {
  const float* Hs  = (const float*)d_in[0];
  const float* Hp  = (const float*)d_in[1];
  const float* Hd  = (const float*)d_in[2];
  const float* pos = (const float*)d_in[3];
  const float* w_s = (const float*)d_in[4];
  const float* w_p = (const float*)d_in[5];
  const float* w_d = (const float*)d_in[6];
  const int* esrc  = (const int*)d_in[7];
  const int* edst  = (const int*)d_in[8];
  const int* gids  = (const int*)d_in[9];

  const int N = in_sizes[0] / 4;
  const int E = in_sizes[7];
  const int G = out_size - 10 * E;   // num_graphs (out = [G graph sums | E*10 emb])

  float* out = (float*)d_out;
  float* emb = out + G;
  float* C   = (float*)d_ws;         // 152 floats

  const int zb = (G + 127) / 128;
  cg_setup_kernel<<<2 + zb, 128, 0, stream>>>(C, out, G);

  const int ab = (N + 255) / 256;
  const int eb = (E + 255) / 256;
  fused_kernel<<<ab + eb, 256, 0, stream>>>(Hs, Hp, Hd, w_s, w_p, w_d, gids, C, out,
                                            esrc, edst, pos, emb, N, E, ab);
}